// RegimeFeatureExtractor_43671227466307
// MI455X (gfx1250) — compile-verified
//
#include <hip/hip_runtime.h>
#include <stdint.h>

// RegimeFeatureExtractor for MI455X (gfx1250).
// Memory-bound gather+reduce: 132 MB in / 2.75 MB out -> ~5.8us floor at 23.3 TB/s.
// CDNA5 path: global_load_async_to_lds_b32 per-lane gathers (ASYNCcnt-tracked
// DMA into LDS, no VGPR round trip), GVS addressing (SGPR base + 32-bit lane
// offset), s_wait_asynccnt + workgroup barrier before the LDS compute phase.

#define T_LEN   63
#define NFEAT   16
#define ROWB    (NFEAT * 4)        // 64 bytes per time row
#define NB      128                // batches per block == threads per block
#define STRIDE  89                 // odd float stride -> bank-conflict-free
// LDS slots per batch: r[0..62] at 0..62, volf[42..62] at 63..83,
// rsi[62] at 84, momf[62] at 85  (86 used, 89 allocated)

// GVS form: copies 4 bytes from global address sbase plus 32-bit offset goff to LDS address lds_off; bumps ASYNCcnt
__device__ __forceinline__ void async_gather_b32(uint32_t lds_off,
                                                 uint32_t goff,
                                                 unsigned long long sbase) {
  asm volatile("global_load_async_to_lds_b32 %0, %1, %2"
               :: "v"(lds_off), "v"(goff), "s"(sbase) : "memory");
}

__global__ __launch_bounds__(NB) void regime_features_kernel(
    const float* __restrict__ x, float* __restrict__ out, int B) {
  __shared__ float sm[NB * STRIDE];

  const int tid = threadIdx.x;
  int b = blockIdx.x * NB + tid;
  if (b >= B) b = B - 1;  // duplicate last batch (benign same-value writes)

  // ---------------- Phase 1: async gather needed columns into LDS ----------
  const unsigned long long sbase = (unsigned long long)(uintptr_t)x;
  const uint32_t bbase = (uint32_t)b * (uint32_t)(T_LEN * ROWB);
  // Low 32 bits of the generic shared pointer == LDS byte offset.
  const uint32_t ls = (uint32_t)(uintptr_t)(&sm[tid * STRIDE]);

  {
    uint32_t go = bbase;          // x[b,t,0] : r column
    uint32_t lo = ls;
    for (int t = 0; t < T_LEN; ++t) {
      async_gather_b32(lo, go, sbase);
      go += ROWB; lo += 4u;
    }
  }
  {
    uint32_t go = bbase + 42u * ROWB + 40u;   // x[b,t,10], t=42..62 : volf
    uint32_t lo = ls + 4u * 63u;
    for (int t = 0; t < 21; ++t) {
      async_gather_b32(lo, go, sbase);
      go += ROWB; lo += 4u;
    }
  }
  async_gather_b32(ls + 4u * 84u, bbase + 62u * ROWB + 4u,  sbase); // rsi[62]
  async_gather_b32(ls + 4u * 85u, bbase + 62u * ROWB + 60u, sbase); // momf[62]

  asm volatile("s_wait_asynccnt 0x0" ::: "memory");
  __syncthreads();

  // ---------------- Phase 2: per-thread statistics from LDS ----------------
  const int rb = tid * STRIDE;        // sm[rb + t] = r[t]
  const int vb = rb + 63;             // sm[vb + i] = volf[42 + i]

  // Pass 1 over r: sum, abs-sum, min, max
  float sum = 0.f, asum = 0.f, mn = 3.402823466e38f, mx = -3.402823466e38f;
#pragma unroll
  for (int t = 0; t < T_LEN; ++t) {
    float c = sm[rb + t];
    sum += c; asum += fabsf(c);
    mn = fminf(mn, c); mx = fmaxf(mx, c);
  }
  float s21 = 0.f;
#pragma unroll
  for (int t = 42; t < T_LEN; ++t) s21 += sm[rb + t];
  float sv = 0.f;
#pragma unroll
  for (int i = 0; i < 21; ++i) sv += sm[vb + i];

  const float r0  = sm[rb + 0];
  const float r42 = sm[rb + 42];
  const float r62 = sm[rb + 62];

  const float mean  = sum * (1.0f / 63.0f);
  const float meanA = (sum - r62) * (1.0f / 62.0f);   // mean of r[0..61]
  const float meanB = (sum - r0)  * (1.0f / 62.0f);   // mean of r[1..62]
  const float m21   = s21 * (1.0f / 21.0f);

  // Pass 2: central moments, autocorr sums, short-window variance
  float varL = 0.f, cub = 0.f, s21v = 0.f;
#pragma unroll
  for (int t = 0; t < T_LEN; ++t) {
    float d = sm[rb + t] - mean;
    varL += d * d; cub += d * d * d;
  }
  float saa = 0.f, sbb = 0.f, sab = 0.f;
#pragma unroll
  for (int t = 0; t < T_LEN - 1; ++t) {
    float da = sm[rb + t]     - meanA;
    float db = sm[rb + t + 1] - meanB;
    saa += da * da; sbb += db * db; sab += da * db;
  }
#pragma unroll
  for (int t = 42; t < T_LEN; ++t) {
    float d = sm[rb + t] - m21;
    s21v += d * d;
  }

  const float vol_long  = sqrtf(varL * (1.0f / 62.0f));
  const float vol_short = sqrtf(s21v * (1.0f / 20.0f));
  const float vol_ratio = vol_short / (vol_long + 1e-8f);
  const float high_vol  = (vol_short > 0.03f) ? 1.f : 0.f;
  const float med_vol   = (vol_short >= 0.01f && vol_short <= 0.03f) ? 1.f : 0.f;
  const float low_vol   = (vol_short < 0.01f) ? 1.f : 0.f;

  const float corr  = sab / sqrtf(saa * sbb);
  const float trend = __builtin_isnan(corr) ? 0.5f : fabsf(corr);

  const float skew = (vol_long < 1e-8f)
      ? 0.f
      : (cub * (1.0f / 63.0f)) / (vol_long * vol_long * vol_long);

  const float mom_s = r62 / (r42 + 1e-8f) - 1.0f;

  // Sliding-window sample stds, two-pass, fully unrolled with immediate
  // LDS offsets: window k covers r[base .. base+21], base = 41 - 5k.
#define WSTD(base)                                                        \
  ({                                                                      \
    float sw_ = 0.f;                                                      \
    _Pragma("unroll") for (int i_ = 0; i_ < 22; ++i_)                     \
        sw_ += sm[rb + (base) + i_];                                      \
    const float mw_ = sw_ * (1.0f / 22.0f);                               \
    float vw_ = 0.f;                                                      \
    _Pragma("unroll") for (int i_ = 0; i_ < 22; ++i_) {                   \
      float d_ = sm[rb + (base) + i_] - mw_;                              \
      vw_ += d_ * d_;                                                     \
    }                                                                     \
    sqrtf(vw_ * (1.0f / 21.0f));                                          \
  })

  const float vs0 = WSTD(41);
  const float vs1 = WSTD(36);
  const float vs2 = WSTD(31);
  const float vs3 = WSTD(26);
  const float vs4 = WSTD(21);
  const float vs5 = WSTD(16);
  const float vs6 = WSTD(11);
  const float vs7 = WSTD(6);
  const float vs8 = WSTD(1);
#undef WSTD

  const float mvs =
      (vs0 + vs1 + vs2 + vs3 + vs4 + vs5 + vs6 + vs7 + vs8) * (1.0f / 9.0f);
  float pvar = 0.f;
  {
    float d;
    d = vs0 - mvs; pvar += d * d;
    d = vs1 - mvs; pvar += d * d;
    d = vs2 - mvs; pvar += d * d;
    d = vs3 - mvs; pvar += d * d;
    d = vs4 - mvs; pvar += d * d;
    d = vs5 - mvs; pvar += d * d;
    d = vs6 - mvs; pvar += d * d;
    d = vs7 - mvs; pvar += d * d;
    d = vs8 - mvs; pvar += d * d;
  }
  const float vol_pers  = sqrtf(pvar * (1.0f / 9.0f));  // ddof=0 over 9
  const float vol_trend = vs0 - vs8;

  // ---------------- Write 21 outputs -----------------
  float* o = out + (size_t)b * 21;
  o[0]  = high_vol;
  o[1]  = med_vol;
  o[2]  = low_vol;
  o[3]  = trend;
  o[4]  = vol_ratio;
  o[5]  = mean;
  o[6]  = asum * (1.0f / 63.0f);
  o[7]  = mx - mn;
  o[8]  = skew;
  o[9]  = mom_s;
  o[10] = sm[rb + 84];            // rsi_current
  o[11] = sv * (1.0f / 21.0f);    // vol_feature_mean
  o[12] = sm[rb + 85];            // momentum_feature_current
  o[13] = 0.0f;                   // relative_return
  o[14] = 1.0f;                   // relative_vol
  o[15] = 1.0f;                   // beta
  o[16] = 0.0f;                   // relative_strength
  o[17] = 0.02f;                  // market_vol_level
  o[18] = vol_pers;
  o[19] = vol_trend;
  o[20] = 0.0f;                   // regime_shift
}

extern "C" void kernel_launch(void* const* d_in, const int* in_sizes, int n_in,
                              void* d_out, int out_size, void* d_ws, size_t ws_size,
                              hipStream_t stream) {
  (void)n_in; (void)d_ws; (void)ws_size; (void)out_size;
  const float* x = (const float*)d_in[0];
  float* out = (float*)d_out;
  const int B = in_sizes[0] / (T_LEN * NFEAT);
  const int blocks = (B + NB - 1) / NB;
  hipLaunchKernelGGL(regime_features_kernel, dim3(blocks), dim3(NB), 0, stream,
                     x, out, B);
}